// ROIPoolingLayer_25005299597626
// MI455X (gfx1250) — compile-verified
//
#include <hip/hip_runtime.h>

// ROI-Align pooling, MI455X (gfx1250, wave32).
//
// Roofline: 205.5 MB fp32 output (NT-stored), feature map 16.8 MB resident in
// 192 MB L2 -> HBM floor ~9 us. Instruction issue is the co-limiter, so:
//   Kernel A: per-ROI interp tables (gy0,gy1,wy / gx0,gx1,wx), 1024 waves.
//   Kernel B: 2-D grid (28 x 1024): roi = blockIdx.y (SGPR, no div-by-196).
//             Wave0 async-copies the ROI's 512B table to LDS
//             (global_load_async_to_lds_b128 + s_wait_asynccnt), all waves
//             broadcast-read it from LDS, readfirstlane -> SALU addressing,
//             saddr-form b128 gathers, NT b128 stores.

typedef __attribute__((ext_vector_type(4))) float v4f;

#define PH   14
#define PW   14
#define FH   128
#define FW   128
#define FC   256
#define NROI 1024
#define CELLS (NROI * PH * PW)     // 200704

__device__ __forceinline__ int rfl_i(int v) {
    return __builtin_amdgcn_readfirstlane(v);
}
__device__ __forceinline__ float rfl_f(float v) {
    return __int_as_float(__builtin_amdgcn_readfirstlane(__float_as_int(v)));
}

// ---------------------------------------------------------------------------
// Kernel A: per-ROI interpolation tables.
// 1024 blocks x 32 threads (one wave per ROI).
// Entry layout per ROI (32 x v4f = 512B):
//   [0..13]  y-entries {gy0, gy1, wy, 0}   (stored as floats)
//   [14..27] x-entries {gx0, gx1, wx, 0}
//   [28..31] zero padding (so kernel B can async-copy with all 32 lanes)
// ---------------------------------------------------------------------------
__global__ __launch_bounds__(32)
void roi_tab_kernel(const float* __restrict__ rois, float* __restrict__ tabg)
{
    const int roi = blockIdx.x;
    const int t   = threadIdx.x;          // 0..31

    const v4f r = *reinterpret_cast<const v4f*>(rois + roi * 4);
    const float x1 = floorf(r.x * (float)FW);
    const float y1 = floorf(r.y * (float)FH);
    const float x2 = floorf((r.x + r.z) * (float)FW);
    const float y2 = floorf((r.y + r.w) * (float)FH);
    const float in_h = y2 - y1;
    const float in_w = x2 - x1;

    const bool  isY  = (t < 14);
    const int   p    = isY ? t : (t - 14);
    const float base = isY ? y1 : x1;
    const float len  = isY ? in_h : in_w;

    float c = ((float)p + 0.5f) * len * (1.0f / 14.0f) - 0.5f;
    c = fminf(fmaxf(c, 0.0f), len - 1.0f);
    const float f0 = floorf(c);
    const float wg = c - f0;

    const int g0  = (int)(base + f0);
    const int lim = (int)(base + len - 1.0f);
    const int g1  = (g0 + 1 < lim) ? g0 + 1 : lim;

    v4f e;
    e.x = (float)g0; e.y = (float)g1; e.z = wg; e.w = 0.0f;
    if (t >= 28) { e.x = 0.0f; e.y = 0.0f; e.z = 0.0f; e.w = 0.0f; }

    reinterpret_cast<v4f*>(tabg)[(size_t)roi * 32 + t] = e;
}

// ---------------------------------------------------------------------------
// Kernel B: main gather/blend. Grid (28, 1024), 224 threads = 7 waves.
// Wave w handles cell blockIdx.x*7 + w of ROI blockIdx.y; each lane does
// 8 channels (2x b128 per corner, 2x NT b128 stores).
// ---------------------------------------------------------------------------
__global__ __launch_bounds__(224)
void roi_align_main_kernel(const float* __restrict__ fm,
                           const float* __restrict__ tabg,
                           float* __restrict__ out)
{
    __shared__ v4f tab[32];               // this ROI's table (512B)

    const int roi  = blockIdx.y;          // SGPR
    const int lane = threadIdx.x & 31;

    // Wave 0: async-copy the ROI table global -> LDS (CDNA5 ASYNCcnt path).
    if (threadIdx.x < 32) {
        unsigned ldsa = (unsigned)(uintptr_t)&tab[lane];  // low 32 bits = LDS offset
        const v4f* g  = reinterpret_cast<const v4f*>(tabg) + (size_t)roi * 32 + lane;
        asm volatile("global_load_async_to_lds_b128 %0, %1, off\n\t"
                     "s_wait_asynccnt 0x0"
                     :: "v"(ldsa), "v"(g) : "memory");
    }
    __syncthreads();

    const int w   = threadIdx.x >> 5;            // wave in block, 0..6
    const int cir = blockIdx.x * 7 + w;          // cell in ROI, 0..195
    const int py  = (cir * 2341) >> 15;          // cir / 14
    const int px  = cir - py * 14;

    // Broadcast reads from LDS (all lanes same address -> conflict-free).
    const v4f ey = tab[py];
    const v4f ex = tab[14 + px];

    // Scalarize the wave-uniform values -> SALU addressing, saddr-form loads.
    const int   gy0 = rfl_i((int)ey.x);
    const int   gy1 = rfl_i((int)ey.y);
    const int   gx0 = rfl_i((int)ex.x);
    const int   gx1 = rfl_i((int)ex.y);
    const float wy  = rfl_f(ey.z);
    const float wx  = rfl_f(ex.z);

    const float w00 = (1.0f - wy) * (1.0f - wx);
    const float w01 = (1.0f - wy) * wx;
    const float w10 = wy * (1.0f - wx);
    const float w11 = wy * wx;

    const int sb00 = (gy0 * FW + gx0) * (FC / 4);   // float4 units, scalar
    const int sb01 = (gy0 * FW + gx1) * (FC / 4);
    const int sb10 = (gy1 * FW + gx0) * (FC / 4);
    const int sb11 = (gy1 * FW + gx1) * (FC / 4);

    const v4f* fm4 = reinterpret_cast<const v4f*>(fm);
    const int  c4  = lane * 2;                      // 2 float4 = 8 channels/lane

    const v4f a00 = fm4[sb00 + c4],     a01 = fm4[sb01 + c4];
    const v4f a10 = fm4[sb10 + c4],     a11 = fm4[sb11 + c4];
    const v4f s00 = fm4[sb00 + c4 + 1], s01 = fm4[sb01 + c4 + 1];
    const v4f s10 = fm4[sb10 + c4 + 1], s11 = fm4[sb11 + c4 + 1];

    const v4f o0 = a00 * w00 + a01 * w01 + a10 * w10 + a11 * w11;
    const v4f o1 = s00 * w00 + s01 * w01 + s10 * w10 + s11 * w11;

    const size_t cell = (size_t)roi * (PH * PW) + cir;
    v4f* out4 = reinterpret_cast<v4f*>(out) + cell * (FC / 4) + c4;
    __builtin_nontemporal_store(o0, out4);
    __builtin_nontemporal_store(o1, out4 + 1);
}

// ---------------------------------------------------------------------------
// Fallback: self-contained single kernel (used only if ws is too small).
// ---------------------------------------------------------------------------
__global__ __launch_bounds__(256)
void roi_align_fallback_kernel(const float* __restrict__ fm,
                               const float* __restrict__ rois,
                               float* __restrict__ out)
{
    const int lane = threadIdx.x & 31;
    const int wave = threadIdx.x >> 5;
    const int cell = blockIdx.x * 8 + wave;

    const int roi = cell / (PH * PW);
    const int rem = cell - roi * (PH * PW);
    const int py  = rem / PW;
    const int px  = rem - py * PW;

    const v4f r = *reinterpret_cast<const v4f*>(rois + roi * 4);
    const float x1 = floorf(r.x * (float)FW);
    const float y1 = floorf(r.y * (float)FH);
    const float x2 = floorf((r.x + r.z) * (float)FW);
    const float y2 = floorf((r.y + r.w) * (float)FH);
    const float in_h = y2 - y1;
    const float in_w = x2 - x1;

    float cy = ((float)py + 0.5f) * in_h * (1.0f / 14.0f) - 0.5f;
    float cx = ((float)px + 0.5f) * in_w * (1.0f / 14.0f) - 0.5f;
    cy = fminf(fmaxf(cy, 0.0f), in_h - 1.0f);
    cx = fminf(fmaxf(cx, 0.0f), in_w - 1.0f);

    const float fy0 = floorf(cy), fx0 = floorf(cx);
    const float wy = cy - fy0,    wx = cx - fx0;
    const int gy0 = (int)(y1 + fy0);
    const int gx0 = (int)(x1 + fx0);
    const int ylim = (int)(y1 + in_h - 1.0f);
    const int xlim = (int)(x1 + in_w - 1.0f);
    const int gy1 = (gy0 + 1 < ylim) ? gy0 + 1 : ylim;
    const int gx1 = (gx0 + 1 < xlim) ? gx0 + 1 : xlim;

    const float w00 = (1.0f - wy) * (1.0f - wx);
    const float w01 = (1.0f - wy) * wx;
    const float w10 = wy * (1.0f - wx);
    const float w11 = wy * wx;

    const v4f* fm4 = reinterpret_cast<const v4f*>(fm);
    const int  c4  = lane * 2;
    const int b00 = (gy0 * FW + gx0) * (FC / 4) + c4;
    const int b01 = (gy0 * FW + gx1) * (FC / 4) + c4;
    const int b10 = (gy1 * FW + gx0) * (FC / 4) + c4;
    const int b11 = (gy1 * FW + gx1) * (FC / 4) + c4;

    const v4f o0 = fm4[b00] * w00 + fm4[b01] * w01 + fm4[b10] * w10 + fm4[b11] * w11;
    const v4f o1 = fm4[b00+1] * w00 + fm4[b01+1] * w01 + fm4[b10+1] * w10 + fm4[b11+1] * w11;

    v4f* out4 = reinterpret_cast<v4f*>(out) + (size_t)cell * (FC / 4) + c4;
    __builtin_nontemporal_store(o0, out4);
    __builtin_nontemporal_store(o1, out4 + 1);
}

extern "C" void kernel_launch(void* const* d_in, const int* in_sizes, int n_in,
                              void* d_out, int out_size, void* d_ws, size_t ws_size,
                              hipStream_t stream) {
    const float* fm   = (const float*)d_in[0];   // (1,128,128,256) f32
    const float* rois = (const float*)d_in[1];   // (1024,4) f32
    float*       out  = (float*)d_out;

    const size_t tab_bytes = (size_t)NROI * 32 * sizeof(v4f);  // 512 KB

    if (ws_size >= tab_bytes) {
        float* tabg = (float*)d_ws;
        roi_tab_kernel<<<NROI, 32, 0, stream>>>(rois, tabg);
        dim3 grid(28, NROI);                       // 28 blocks x 7 waves = 196 cells
        roi_align_main_kernel<<<grid, 224, 0, stream>>>(fm, tabg, out);
    } else {
        roi_align_fallback_kernel<<<CELLS / 8, 256, 0, stream>>>(fm, rois, out);
    }
}